// jslab_56332791054631
// MI455X (gfx1250) — compile-verified
//
#include <hip/hip_runtime.h>
#include <math.h>

typedef __attribute__((ext_vector_type(2))) float v2f;
typedef __attribute__((ext_vector_type(8))) float v8f;

#define GNY 256
#define GNX 256
#define FCg 1e-4f
#define DTg 360.0f

// Zero U[0], V[0] snapshots (harness poisons d_out).
__global__ void slab_zero2(float* __restrict__ a, float* __restrict__ b, int n) {
    int i = blockIdx.x * blockDim.x + threadIdx.x;
    if (i < n) { a[i] = 0.0f; b[i] = 0.0f; }
}

// One Euler substep: U,V -> U',V' with fused conv1(relu)->conv2(relu) dissipation.
// 16x16 output tile per block, 8 wave32s. K-packings are chosen so the lane-half
// (K sub-block) dependence is a constant LDS address shift: after full unroll all
// fragment loads are immediate-offset ds_loads off one per-lane base register.
__global__ __launch_bounds__(256) void slab_step(
    const float* __restrict__ Uin, const float* __restrict__ Vin,
    float* __restrict__ Uout, float* __restrict__ Vout,
    const float* __restrict__ tax0, const float* __restrict__ tax1,
    const float* __restrict__ tay0, const float* __restrict__ tay1,
    const float* __restrict__ K0,
    const float* __restrict__ w1, const float* __restrict__ b1,
    const float* __restrict__ w2, const float* __restrict__ b2,
    float aa)
{
    __shared__ float xin[2][21][20];    // input + 2-halo; row 20 is a zero pad row
    __shared__ float hid[16][18][18];   // conv1 output with 1-halo
    __shared__ float w1A[20][16];       // conv1 A: q = tap + 10*ic, tap 9 zeroed
    __shared__ float w2A[144][16];      // conv2 A: q = C + 72*half, rows m>=2 zeroed
    __shared__ float b1s[16];
    __shared__ float b2s[2];

    const int tid  = threadIdx.x;
    const int wave = tid >> 5;
    const int lane = tid & 31;
    const int half = lane >> 4;   // K sub-block of the wave
    const int nl   = lane & 15;   // M (A rows) / N (B,C,D cols) index
    const int bx = blockIdx.x, by = blockIdx.y;

    const float K  = expf(K0[0]);       // scalar load hides under stages 0/1
    const float ab = 1.0f - aa;

    // ---- stage 0: padded A layouts + biases + input halo into LDS ----
    for (int i = tid; i < 20 * 16; i += 256) {      // w1A[q][m], q = tap + 10*ic
        int q = i >> 4, m = i & 15;
        int ic = q / 10, tap = q - 10 * ic;
        float v = 0.0f;
        if (tap < 9) {
            int dy = tap / 3, dx = tap - 3 * dy;
            v = w1[((m * 2 + ic) * 3 + dy) * 3 + dx];
        }
        w1A[q][m] = v;
    }
    for (int i = tid; i < 144 * 16; i += 256) {     // w2A[q][m], q = ic*9 + tap
        int q = i >> 4, m = i & 15;
        float v = 0.0f;
        if (m < 2) {
            int ic = q / 9, tap = q - 9 * ic;
            int dy = tap / 3, dx = tap - 3 * dy;
            v = w2[((m * 16 + ic) * 3 + dy) * 3 + dx];
        }
        w2A[q][m] = v;
    }
    if (tid < 16) b1s[tid] = b1[tid];
    if (tid < 2)  b2s[tid] = b2[tid];

    for (int i = tid; i < 2 * 21 * 20; i += 256) {  // includes zero pad row 20
        int ch  = i / 420;
        int rem = i - ch * 420;
        int rr  = rem / 20, cc = rem - rr * 20;
        int gr = by * 16 + rr - 2, gc = bx * 16 + cc - 2;
        float v = 0.0f;
        if (rr < 20 && gr >= 0 && gr < GNY && gc >= 0 && gc < GNX) {
            const float* src = ch ? Vin : Uin;
            v = src[gr * GNX + gc];
        }
        xin[ch][rr][cc] = v;
    }
    __syncthreads();

    // ---- stage 1: conv1 (2->16ch) via WMMA f32 16x16x4 ----
    // q = (2*blk + i) + 10*half  =>  ic == half (folded into base pointer),
    // tap == 2*blk+i compile-time (tap 9 hits the zero pad row & zero A column).
    const float* xp  = &xin[0][0][0] + half * 420;          // channel = half
    const float* w1p = &w1A[0][0] + half * 160 + nl;
    v2f a1[5];
#pragma unroll
    for (int blk = 0; blk < 5; ++blk) {
        a1[blk][0] = w1p[(2 * blk + 0) * 16];
        a1[blk][1] = w1p[(2 * blk + 1) * 16];
    }
    // Tasks: 18 hidden rows x 2 overlapping column groups (c0 = 0, 2).
    // Group 1 recomputes cols 2..15 bitwise-identically (benign duplicate stores).
    for (int task = wave; task < 36; task += 8) {
        int r  = task >> 1;
        int c0 = (task & 1) << 1;
        int xbase = r * 20 + c0 + nl;
        v8f acc = {0.f, 0.f, 0.f, 0.f, 0.f, 0.f, 0.f, 0.f};
#pragma unroll
        for (int blk = 0; blk < 5; ++blk) {
            v2f b;
#pragma unroll
            for (int i = 0; i < 2; ++i) {
                int tap = 2 * blk + i;              // compile-time
                int dy = tap / 3, dx = tap - 3 * dy;
                b[i] = xp[xbase + dy * 20 + dx];    // immediate ds offset
            }
            acc = __builtin_amdgcn_wmma_f32_16x16x4_f32(
                false, a1[blk], false, b, (short)0, acc, false, false);
        }
#pragma unroll
        for (int v = 0; v < 8; ++v) {
            int m = v + 8 * half;                   // D: vgpr v holds row v+8*half
            float h = acc[v] + b1s[m];
            hid[m][r][c0 + nl] = h > 0.0f ? h : 0.0f;
        }
    }
    __syncthreads();

    // ---- stage 2: conv2 (16->2ch) via WMMA + Euler update ----
    // q = C + 72*half, C = 2*blk+i. 72 % 9 == 0 => ic = C/9 + 8*half,
    // tap = C % 9: half contributes a constant base shift, C is compile-time.
    {
        const int r0 = wave;                        // rows r0 and r0+8
        const float* hp  = &hid[0][0][0] + half * (8 * 324);
        const float* w2p = &w2A[0][0] + half * (72 * 16) + nl;
        const int hbA = r0 * 18 + nl;
        const int hbB = (r0 + 8) * 18 + nl;
        v8f acc0 = {0.f, 0.f, 0.f, 0.f, 0.f, 0.f, 0.f, 0.f};
        v8f acc1 = {0.f, 0.f, 0.f, 0.f, 0.f, 0.f, 0.f, 0.f};
#pragma unroll
        for (int blk = 0; blk < 36; ++blk) {
            v2f a, bA, bB;
#pragma unroll
            for (int i = 0; i < 2; ++i) {
                int C   = 2 * blk + i;              // compile-time after unroll
                int icl = C / 9;
                int tap = C - 9 * icl;
                int dy  = tap / 3, dx = tap - 3 * dy;
                int off = icl * 324 + dy * 18 + dx; // compile-time
                a[i]  = w2p[C * 16];                // immediate ds offset
                bA[i] = hp[off + hbA];              // immediate ds offset
                bB[i] = hp[off + hbB];
            }
            acc0 = __builtin_amdgcn_wmma_f32_16x16x4_f32(
                false, a, false, bA, (short)0, acc0, false, false);
            acc1 = __builtin_amdgcn_wmma_f32_16x16x4_f32(
                false, a, false, bB, (short)0, acc1, false, false);
        }
        // rows 0,1 of each D live in acc[0],acc[1] of lanes 0..15
        if (half == 0) {
#pragma unroll
            for (int s = 0; s < 2; ++s) {
                int r = r0 + 8 * s;
                float d0 = (s ? acc1[0] : acc0[0]) + b2s[0];
                float d1 = (s ? acc1[1] : acc0[1]) + b2s[1];
                float dis0 = d0 > 0.0f ? d0 : 0.0f;
                float dis1 = d1 > 0.0f ? d1 : 0.0f;
                int gr = by * 16 + r, gc = bx * 16 + nl;
                int idx = gr * GNX + gc;
                float U = xin[0][r + 2][nl + 2];
                float V = xin[1][r + 2][nl + 2];
                float tax = ab * tax0[idx] + aa * tax1[idx];
                float tay = ab * tay0[idx] + aa * tay1[idx];
                float dU =  FCg * V + K * tax - dis0;
                float dV = -FCg * U + K * tay - dis1;
                Uout[idx] = U + DTg * dU;
                Vout[idx] = V + DTg * dV;
            }
        }
    }
}

extern "C" void kernel_launch(void* const* d_in, const int* in_sizes, int n_in,
                              void* d_out, int out_size, void* d_ws, size_t ws_size,
                              hipStream_t stream) {
    (void)in_sizes; (void)n_in; (void)ws_size;
    const float* TAx = (const float*)d_in[0];
    const float* TAy = (const float*)d_in[1];
    const float* K0  = (const float*)d_in[2];
    const float* w1  = (const float*)d_in[3];
    const float* b1  = (const float*)d_in[4];
    const float* w2  = (const float*)d_in[5];
    const float* b2  = (const float*)d_in[6];
    // d_in[7], d_in[8] are t0,t1 (device scalars, not host-readable under graph
    // capture); Nforcing is recovered from out_size instead.

    const int N  = GNY * GNX;
    const int Nf = out_size / (2 * N);      // 24
    float* outU = (float*)d_out;
    float* outV = outU + (size_t)Nf * N;

    float* ws  = (float*)d_ws;              // 4 x 256KB ping-pong state
    float* wU0 = ws;
    float* wV0 = ws + N;
    float* wU1 = ws + 2 * (size_t)N;
    float* wV1 = ws + 3 * (size_t)N;

    // U[0] = V[0] = 0
    slab_zero2<<<(N + 255) / 256, 256, 0, stream>>>(outU, outV, N);

    dim3 grid(GNX / 16, GNY / 16), block(256);
    // iout = Nf-1 is integrated but dropped by the reference -> skip it.
    for (int iout = 0; iout < Nf - 1; ++iout) {
        for (int iin = 0; iin < 10; ++iin) {
            const float *Ui, *Vi;
            float *Uo, *Vo;
            if (iin == 0) { Ui = outU + (size_t)iout * N; Vi = outV + (size_t)iout * N; }
            else { int p = (iin - 1) & 1; Ui = p ? wU1 : wU0; Vi = p ? wV1 : wV0; }
            if (iin == 9) { Uo = outU + (size_t)(iout + 1) * N; Vo = outV + (size_t)(iout + 1) * N; }
            else { int p = iin & 1; Uo = p ? wU1 : wU0; Vo = p ? wV1 : wV0; }
            float aa = (float)iin * 0.1f;   // itf = iout, itsup = iout+1 (iout <= Nf-2)
            slab_step<<<grid, block, 0, stream>>>(
                Ui, Vi, Uo, Vo,
                TAx + (size_t)iout * N, TAx + (size_t)(iout + 1) * N,
                TAy + (size_t)iout * N, TAy + (size_t)(iout + 1) * N,
                K0, w1, b1, w2, b2, aa);
        }
    }
}